// ConstrainLoss_50190987821156
// MI455X (gfx1250) — compile-verified
//
#include <hip/hip_runtime.h>
#include <hip/hip_bf16.h>

// ---------------------------------------------------------------------------
// ConstrainLoss for MI455X (gfx1250, wave32)
//   input : (B=256, H=13, W=13, C=1024) fp32, C contiguous
//   output: single fp32 scalar
//
// Pass 1: per-(b,h,w) softmax constants  a = max + log(sum exp(x-max))
// Pass 2: per-(b, 16-channel tile) the five weighted spatial sums
//         (S, Sx, Sy, Sxx, Syy) via V_WMMA_F32_16X16X4_F32. Weights and row
//         constants are pre-packed per block into one LDS float4 per
//         (spatial-pair, weight-row) so the hot loop is one ds_load_b128 +
//         two global_load_b32 + two v_exp + one WMMA.
// ---------------------------------------------------------------------------

#define GRID_HW   13          // GRID-1
#define SPATIAL   169         // 13*13
#define NJ        85          // ceil(SPATIAL/2): spatial pairs (2j, 2j+1)
#define CHANNELS  1024
#define BATCH     256
#define Z_CONST   17.0794684453471326f   // exp(log(2*pi)+1) = 2*pi*e
#define EPS_CONST 1e-6f
#define INV169    (1.0f / 169.0f)
#define INV_BC    (1.0f / (256.0f * 1024.0f))

typedef float v2f __attribute__((ext_vector_type(2)));
typedef float v8f __attribute__((ext_vector_type(8)));

__global__ void zero_out_kernel(float* out) { out[0] = 0.0f; }

// ---------------------------------------------------------------------------
// Pass 1: one 256-thread block per softmax row (B*169 rows of 1024 floats).
// ---------------------------------------------------------------------------
__global__ void __launch_bounds__(256)
softmax_rowconst_kernel(const float* __restrict__ x, float* __restrict__ rowconst) {
    const int row  = blockIdx.x;
    const int tid  = threadIdx.x;
    const int wave = tid >> 5;
    const int lane = tid & 31;

    const float4 v = reinterpret_cast<const float4*>(x + (size_t)row * CHANNELS)[tid];

    float m = fmaxf(fmaxf(v.x, v.y), fmaxf(v.z, v.w));
#pragma unroll
    for (int d = 16; d >= 1; d >>= 1) m = fmaxf(m, __shfl_xor(m, d, 32));

    __shared__ float red[8];
    if (lane == 0) red[wave] = m;
    __syncthreads();
    float bm = red[0];
#pragma unroll
    for (int i = 1; i < 8; ++i) bm = fmaxf(bm, red[i]);

    float e = __expf(v.x - bm) + __expf(v.y - bm) +
              __expf(v.z - bm) + __expf(v.w - bm);
#pragma unroll
    for (int d = 16; d >= 1; d >>= 1) e += __shfl_xor(e, d, 32);

    __syncthreads();                 // red[] reuse
    if (lane == 0) red[wave] = e;
    __syncthreads();

    if (tid == 0) {
        float s = red[0];
#pragma unroll
        for (int i = 1; i < 8; ++i) s += red[i];
        rowconst[row] = bm + __logf(s);
    }
}

// weight row m at flat spatial cell s (one-time setup cost only)
__device__ __forceinline__ float moment_weight(int m, int s) {
    const int   h  = s / GRID_HW;
    const int   w  = s - h * GRID_HW;
    const float xv = (float)(h + 1);
    const float yv = (float)(w + 1);
    float r = 0.0f;
    if      (m == 0) r = 1.0f;
    else if (m == 1) r = xv;
    else if (m == 2) r = yv;
    else if (m == 3) r = xv * xv;
    else if (m == 4) r = yv * yv;
    return r;
}

// ---------------------------------------------------------------------------
// Pass 2: 128 threads = 4 waves per block; wave handles one (b, 16-ch tile).
// grid = 256 b * 16 groups = 4096 blocks; group*4 + wave = channel tile 0..63.
// ---------------------------------------------------------------------------
__global__ void __launch_bounds__(128)
spatial_moment_kernel(const float* __restrict__ x,
                      const float* __restrict__ rowconst,
                      float* __restrict__ out) {
    // tab[j*16 + n] = { w(n,2j), w(n,2j+1), a[2j], a[2j+1] }
    __shared__ float4 tab[NJ * 16];
    __shared__ float  blockAcc;

    const int b    = blockIdx.x >> 4;
    const int grp  = blockIdx.x & 15;
    const int tid  = threadIdx.x;
    const int wave = tid >> 5;
    const int lane = tid & 31;

    if (tid == 0) blockAcc = 0.0f;

    // One-time per-block table build (1360 entries over 128 threads).
    const float* rc = rowconst + b * SPATIAL;
    for (int idx = tid; idx < NJ * 16; idx += 128) {
        const int j  = idx >> 4;
        const int n  = idx & 15;
        const int s0 = 2 * j;
        const int s1 = s0 + 1;
        float4 e;
        e.x = moment_weight(n, s0);
        e.y = (s1 < SPATIAL) ? moment_weight(n, s1) : 0.0f;  // kills cell 169
        e.z = rc[s0];
        e.w = rc[(s1 < SPATIAL) ? s1 : (SPATIAL - 1)];       // finite filler
        tab[idx] = e;
    }
    __syncthreads();

    const int ctile = grp * 4 + wave;      // 0..63
    const int c0    = ctile * 16;
    const int n     = lane & 31 & 15;      // channel within tile == weight row
    const int half  = lane >> 4;           // K half (0 -> K=0,1 ; 1 -> K=2,3)

    const float*  lanebase = x + (size_t)b * SPATIAL * CHANNELS
                               + (size_t)(2 * half) * CHANNELS + c0 + n;
    const float4* lanetab  = &tab[half * 16 + n];   // j = 2*it + half

    v8f acc = {};

#pragma unroll 6
    for (int it = 0; it < 42; ++it) {
        const float  x0 = lanebase[0];           // spatial s0 = 4*it + 2*half
        const float  x1 = lanebase[CHANNELS];    // spatial s0 + 1
        const float4 e  = *lanetab;              // one ds_load_b128

        v2f bv;                                  // B: softmax values
        bv.x = __expf(x0 - e.z);
        bv.y = __expf(x1 - e.w);
        v2f av;                                  // A: precomputed weights
        av.x = e.x;
        av.y = e.y;

        acc = __builtin_amdgcn_wmma_f32_16x16x4_f32(
                  false, av, false, bv, (short)0, acc, false, false);

        lanebase += 4 * CHANNELS;
        lanetab  += 32;                          // advance j by 2
    }

    // Tail: spatial cell 168 (j = 84). Only K slot 0 of half 0 is real; all
    // other slots zeroed through the A weight with finite B values.
    {
        const float4 e  = tab[84 * 16 + n];
        const float* tb = x + (size_t)b * SPATIAL * CHANNELS
                            + (size_t)(SPATIAL - 1) * CHANNELS + c0 + n;
        const float  fT = __expf(tb[0] - e.z);
        v2f av, bv;
        av.x = (half == 0) ? e.x : 0.0f;  av.y = 0.0f;
        bv.x = fT;                         bv.y = 0.0f;
        acc = __builtin_amdgcn_wmma_f32_16x16x4_f32(
                  false, av, false, bv, (short)0, acc, false, false);
    }

    // Lanes 0-15: acc[0]=S, acc[1]=Sx, acc[2]=Sy, acc[3]=Sxx, acc[4]=Syy for
    // channel c0+n. Lanes 16-31 hold the all-zero weight rows M=8..12.
    const float S   = acc[0];
    const float Sx  = acc[1];
    const float Sy  = acc[2];
    const float Sxx = acc[3];
    const float Syy = acc[4];

    const float s     = S + EPS_CONST;
    const float inv_s = 1.0f / s;
    const float mx    = Sx * inv_s;
    const float my    = Sy * inv_s;
    const float xvar  = (Sxx - 2.0f * mx * Sx + mx * mx * S) * inv_s * INV169;
    const float yvar  = (Syy - 2.0f * my * Sy + my * my * S) * inv_s * INV169;
    const float t     = xvar + yvar;
    float val = t * t * Z_CONST * INV_BC;   // lanes 16-31: t == 0 -> val == 0

#pragma unroll
    for (int d = 16; d >= 1; d >>= 1) val += __shfl_xor(val, d, 32);

    if (lane == 0) atomicAdd(&blockAcc, val);
    __syncthreads();
    if (tid == 0) atomicAdd(out, blockAcc);
}

extern "C" void kernel_launch(void* const* d_in, const int* in_sizes, int n_in,
                              void* d_out, int out_size, void* d_ws, size_t ws_size,
                              hipStream_t stream) {
    const float* x   = (const float*)d_in[0];
    float*       out = (float*)d_out;
    float*       rc  = (float*)d_ws;       // B*169 floats = 173 KB of scratch

    zero_out_kernel<<<1, 1, 0, stream>>>(out);
    softmax_rowconst_kernel<<<BATCH * SPATIAL, 256, 0, stream>>>(x, rc);
    spatial_moment_kernel<<<BATCH * 16, 128, 0, stream>>>(x, rc, out);
}